// GCNLayer_67920612819495
// MI455X (gfx1250) — compile-verified
//
#include <hip/hip_runtime.h>
#include <hip/hip_bf16.h>

typedef __attribute__((ext_vector_type(2))) float v2f;
typedef __attribute__((ext_vector_type(8))) float v8f;

#define D_FEAT 128
#define BN_EPS 1e-5f

// ---------------------------------------------------------------------------
// K1: deg = 1.0 (self-loop), zero BN accumulators (sums[128], sumsq[128])
// ---------------------------------------------------------------------------
__global__ void gcn_init_kernel(float* __restrict__ deg, float* __restrict__ stats, int N) {
    int i = blockIdx.x * blockDim.x + threadIdx.x;
    if (i < N) deg[i] = 1.0f;
    if (i < 256) stats[i] = 0.0f;
}

// ---------------------------------------------------------------------------
// K2: in-degree via f32 atomics over dst indices
// ---------------------------------------------------------------------------
__global__ void gcn_degree_kernel(const int* __restrict__ dst, float* __restrict__ deg, int E) {
    int e = blockIdx.x * blockDim.x + threadIdx.x;
    if (e < E) atomicAdd(&deg[dst[e]], 1.0f);
}

// ---------------------------------------------------------------------------
// K3: dinv = rsqrt(deg) in place
// ---------------------------------------------------------------------------
__global__ void gcn_rsqrt_kernel(float* __restrict__ deg, int N) {
    int i = blockIdx.x * blockDim.x + threadIdx.x;
    if (i < N) {
        float d = deg[i];
        deg[i] = (d > 0.0f) ? rsqrtf(d) : 0.0f;
    }
}

// ---------------------------------------------------------------------------
// K4: h = x @ W^T via V_WMMA_F32_16X16X4_F32, fused self-loop agg init:
//     agg[i] = dinv[i]^2 * h[i]
// One wave computes one 16x16 output tile; 8 waves/block cover all 128 cols
// for a 16-row stripe. K = 128 -> 32 WMMA(K=4) steps.
//
// A 16x4 f32 layout (ISA 7.12.2): lanes 0-15 M=lane, VGPR0=K0/VGPR1=K1;
// lanes 16-31 M=lane-16, VGPR0=K2/VGPR1=K3 -> per-lane float2 at k + 2*(lane>>4).
// B 4x16 mirrored: lane picks col j0+(lane&15), float2 of W[col][k + 2*(lane>>4)].
// C/D: VGPR r -> row = r + 8*(lane>>4), col = lane&15.
// ---------------------------------------------------------------------------
__global__ __launch_bounds__(256) void gcn_gemm_wmma_kernel(
        const float* __restrict__ x, const float* __restrict__ W,
        const float* __restrict__ dinv,
        float* __restrict__ h, float* __restrict__ agg, int N) {
    const int lane = threadIdx.x & 31;
    const int wv   = threadIdx.x >> 5;        // 0..7 : column tile
    const int m0   = blockIdx.x * 16;
    const int j0   = wv * 16;
    const int half = lane >> 4;               // 0 or 1
    const int l15  = lane & 15;

    const float* xrow = x + (size_t)(m0 + l15) * D_FEAT + half * 2;
    const float* wrow = W + (size_t)(j0 + l15) * D_FEAT + half * 2;

    v8f c = {};
#pragma unroll
    for (int k0 = 0; k0 < D_FEAT; k0 += 4) {
        v2f a = *(const v2f*)(xrow + k0);
        v2f b = *(const v2f*)(wrow + k0);
        c = __builtin_amdgcn_wmma_f32_16x16x4_f32(false, a, false, b,
                                                  (short)0, c, false, false);
    }

    const int rbase = m0 + half * 8;
    const int col   = j0 + l15;
#pragma unroll
    for (int r = 0; r < 8; ++r) {
        int row = rbase + r;
        float v  = c[r];
        float di = dinv[row];
        h[(size_t)row * D_FEAT + col]   = v;
        agg[(size_t)row * D_FEAT + col] = di * di * v;
    }
}

// Scalar tail for rows not covered by 16-row WMMA stripes (N % 16 != 0).
__global__ void gcn_gemm_tail_kernel(const float* __restrict__ x,
                                     const float* __restrict__ W,
                                     const float* __restrict__ dinv,
                                     float* __restrict__ h, float* __restrict__ agg,
                                     int row0, int N) {
    int idx = blockIdx.x * blockDim.x + threadIdx.x;
    int row = row0 + idx / D_FEAT;
    int col = idx % D_FEAT;
    if (row >= N) return;
    const float* xr = x + (size_t)row * D_FEAT;
    const float* wr = W + (size_t)col * D_FEAT;
    float acc = 0.0f;
    for (int k = 0; k < D_FEAT; ++k) acc = fmaf(xr[k], wr[k], acc);
    float di = dinv[row];
    h[(size_t)row * D_FEAT + col]   = acc;
    agg[(size_t)row * D_FEAT + col] = di * di * acc;
}

// ---------------------------------------------------------------------------
// K5: edge scatter. One wave per edge: lane loads float4 of h[src], does
// 4 global f32 atomic adds into agg[dst] (128 channels / 32 lanes).
// ---------------------------------------------------------------------------
__global__ __launch_bounds__(256) void gcn_scatter_kernel(
        const int* __restrict__ ei, const float* __restrict__ dinv,
        const float* __restrict__ h, float* __restrict__ agg, int E) {
    int lane = threadIdx.x & 31;
    int e = (int)((blockIdx.x * blockDim.x + threadIdx.x) >> 5);
    if (e >= E) return;
    int src = ei[e];
    int dst = ei[E + e];
    float coef = dinv[src] * dinv[dst];
    float4 hv = ((const float4*)(h + (size_t)src * D_FEAT))[lane];
    float* ap = agg + (size_t)dst * D_FEAT + lane * 4;
    atomicAdd(ap + 0, coef * hv.x);
    atomicAdd(ap + 1, coef * hv.y);
    atomicAdd(ap + 2, coef * hv.z);
    atomicAdd(ap + 3, coef * hv.w);
}

// ---------------------------------------------------------------------------
// K6: per-channel sum / sumsq (thread = channel, block-strided rows,
// coalesced loads, one atomic pair per block per channel)
// ---------------------------------------------------------------------------
__global__ __launch_bounds__(128) void gcn_stats_kernel(
        const float* __restrict__ agg, float* __restrict__ sums,
        float* __restrict__ sumsq, int N) {
    int c = threadIdx.x; // 0..127
    float s = 0.0f, sq = 0.0f;
    for (int r = blockIdx.x; r < N; r += gridDim.x) {
        float v = agg[(size_t)r * D_FEAT + c];
        s += v;
        sq = fmaf(v, v, sq);
    }
    atomicAdd(&sums[c], s);
    atomicAdd(&sumsq[c], sq);
}

// K6b: fold mean/var/gamma/beta into per-channel scale & shift
__global__ void gcn_bnparam_kernel(const float* __restrict__ sums,
                                   const float* __restrict__ sumsq,
                                   const float* __restrict__ gamma,
                                   const float* __restrict__ beta,
                                   float* __restrict__ scale,
                                   float* __restrict__ shift, float invN) {
    int c = threadIdx.x;
    float mean = sums[c] * invN;
    float var  = fmaf(-mean, mean, sumsq[c] * invN); // biased variance (jnp.var)
    float sc   = gamma[c] * rsqrtf(var + BN_EPS);
    scale[c] = sc;
    shift[c] = fmaf(-mean, sc, beta[c]);
}

// ---------------------------------------------------------------------------
// K7: out = relu(agg*scale + shift) + x   (float4 vectorized)
// ---------------------------------------------------------------------------
__global__ __launch_bounds__(256) void gcn_finalize_kernel(
        const float4* __restrict__ agg, const float4* __restrict__ x,
        const float4* __restrict__ scale, const float4* __restrict__ shift,
        float4* __restrict__ out, int total4) {
    int i = blockIdx.x * blockDim.x + threadIdx.x;
    if (i >= total4) return;
    int c = i & (D_FEAT / 4 - 1); // channel group 0..31
    float4 a  = agg[i];
    float4 xv = x[i];
    float4 sc = scale[c];
    float4 sh = shift[c];
    float4 o;
    o.x = fmaxf(fmaf(a.x, sc.x, sh.x), 0.0f) + xv.x;
    o.y = fmaxf(fmaf(a.y, sc.y, sh.y), 0.0f) + xv.y;
    o.z = fmaxf(fmaf(a.z, sc.z, sh.z), 0.0f) + xv.z;
    o.w = fmaxf(fmaf(a.w, sc.w, sh.w), 0.0f) + xv.w;
    out[i] = o;
}

// ---------------------------------------------------------------------------
extern "C" void kernel_launch(void* const* d_in, const int* in_sizes, int n_in,
                              void* d_out, int out_size, void* d_ws, size_t ws_size,
                              hipStream_t stream) {
    const float* x     = (const float*)d_in[0];
    const int*   ei    = (const int*)d_in[1];   // [2, E] (src row, then dst row)
    const float* W     = (const float*)d_in[2];
    // d_in[3] = b: added before BatchNorm -> cancels exactly; unused.
    const float* gamma = (const float*)d_in[4];
    const float* beta  = (const float*)d_in[5];

    const int N = in_sizes[0] / D_FEAT;
    const int E = in_sizes[1] / 2;

    float* ws    = (float*)d_ws;
    float* h     = ws;                               // N*128
    float* agg   = h + (size_t)N * D_FEAT;           // N*128
    float* deg   = agg + (size_t)N * D_FEAT;         // N (becomes dinv in place)
    float* stats = deg + (((size_t)N + 3) & ~(size_t)3);
    float* sums  = stats;        // 128
    float* sumsq = stats + 128;  // 128
    float* scale = stats + 256;  // 128
    float* shift = stats + 384;  // 128
    float* out   = (float*)d_out;

    const int T = 256;

    // 1) init deg=1.0, zero BN accumulators
    gcn_init_kernel<<<(N + T - 1) / T, T, 0, stream>>>(deg, stats, N);

    // 2) degree atomics over dst
    gcn_degree_kernel<<<(E + T - 1) / T, T, 0, stream>>>(ei + E, deg, E);

    // 3) dinv = rsqrt(deg)
    gcn_rsqrt_kernel<<<(N + T - 1) / T, T, 0, stream>>>(deg, N);

    // 4) WMMA GEMM + self-loop agg init
    int nStripes = N / 16;
    if (nStripes > 0)
        gcn_gemm_wmma_kernel<<<nStripes, 256, 0, stream>>>(x, W, deg, h, agg, N);
    int tailRows = N - nStripes * 16;
    if (tailRows > 0) {
        int tailElems = tailRows * D_FEAT;
        gcn_gemm_tail_kernel<<<(tailElems + T - 1) / T, T, 0, stream>>>(
            x, W, deg, h, agg, nStripes * 16, N);
    }

    // 5) edge scatter (wave per edge, 8 edges per 256-thread block)
    gcn_scatter_kernel<<<(E + 7) / 8, 256, 0, stream>>>(ei, deg, h, agg, E);

    // 6) BN statistics + fold into scale/shift
    gcn_stats_kernel<<<512, 128, 0, stream>>>(agg, sums, sumsq, N);
    gcn_bnparam_kernel<<<1, 128, 0, stream>>>(sums, sumsq, gamma, beta,
                                              scale, shift, 1.0f / (float)N);

    // 7) BN apply + ReLU + residual
    int total4 = N * (D_FEAT / 4);
    gcn_finalize_kernel<<<(total4 + T - 1) / T, T, 0, stream>>>(
        (const float4*)agg, (const float4*)x, (const float4*)scale,
        (const float4*)shift, (float4*)out, total4);
}